// KFLoss_18047452578536
// MI455X (gfx1250) — compile-verified
//
#include <hip/hip_runtime.h>
#include <math.h>

#define EPS        1e-06f
#define BLOCK      256
#define TILE_BOXES 512                       // boxes per stage
#define TILE_FLTS  (TILE_BOXES * 5)          // 2560 floats per input per stage
#define STAGE_FLTS (TILE_FLTS * 2)           // 5120 floats (pred then target)
#define STAGE_BYTES (STAGE_FLTS * 4)         // 20480 B
#define CHUNKS     (STAGE_BYTES / 16)        // 1280 16-byte chunks per stage
#define CHUNKS_PER_THREAD (CHUNKS / BLOCK)   // exactly 5
#define NBLOCKS    1024

// ---- CDNA5 async global->LDS (ASYNCcnt) ------------------------------------
// GV form: vdst = LDS byte offset (wave-relative), vaddr = 64-bit VGPR pair.
__device__ __forceinline__ void async_load16(unsigned lds_off, const void* gaddr) {
  asm volatile("global_load_async_to_lds_b128 %0, %1, off"
               :: "v"(lds_off), "v"((unsigned long long)gaddr)
               : "memory");
}
__device__ __forceinline__ void wait_async_le5() {
  asm volatile("s_wait_asynccnt 0x5" ::: "memory");
}
__device__ __forceinline__ void wait_async_0() {
  asm volatile("s_wait_asynccnt 0x0" ::: "memory");
}

// Issue exactly CHUNKS_PER_THREAD async loads for one tile stage.
// OOB chunks are CLAMPED to the last valid chunk (keeps per-wave ASYNCcnt
// increments uniform at 5 per stage; garbage lands in slots whose boxes the
// compute loop masks off anyway).
__device__ __forceinline__ void stage_tile(const float* __restrict__ pred,
                                           const float* __restrict__ targ,
                                           long long tileStartBox, long long N,
                                           unsigned ldsBufBase, int tid) {
  const long long floatBase   = tileStartBox * 5;   // per-input float index
  const long long totalFloats = N * 5;              // 20e6
#pragma unroll
  for (int j = 0; j < CHUNKS_PER_THREAD; ++j) {
    const int k   = tid + j * BLOCK;                // 0..1279
    const bool isT = (k >= CHUNKS / 2);             // second half -> target
    const long long kk = isT ? (k - CHUNKS / 2) : k;
    long long f = floatBase + kk * 4;               // chunk start (floats)
    if (f + 4 > totalFloats) f = totalFloats - 4;   // clamp, never skip
    const float* g = (isT ? targ : pred) + f;
    async_load16(ldsBufBase + (unsigned)k * 16u, (const void*)g);
  }
}

// ---- per-pair KFIoU loss ---------------------------------------------------
__device__ __forceinline__ float box_loss(const float* __restrict__ p,
                                          const float* __restrict__ t) {
  // pred gaussian
  float pw = fmaxf(p[2], p[3]), ph = fminf(p[2], p[3]);
  float ps, pc; __sincosf(p[4], &ps, &pc);
  float pw2 = 0.25f * pw * pw, ph2 = 0.25f * ph * ph;
  float P11 = pc * pc * pw2 + ps * ps * ph2 + EPS;
  float P22 = ps * ps * pw2 + pc * pc * ph2 + EPS;
  float P12 = ps * pc * (pw2 - ph2);
  // target gaussian
  float tw = fmaxf(t[2], t[3]), th = fminf(t[2], t[3]);
  float ts, tc; __sincosf(t[4], &ts, &tc);
  float tw2 = 0.25f * tw * tw, th2 = 0.25f * th * th;
  float T11 = tc * tc * tw2 + ts * ts * th2 + EPS;
  float T22 = ts * ts * tw2 + tc * tc * th2 + EPS;
  float T12 = ts * tc * (tw2 - th2);
  // Bhattacharyya distance
  float S11 = P11 + T11, S12 = P12 + T12, S22 = P22 + T22;
  float dx = p[0] - t[0], dy = p[1] - t[1];
  float det_s = S11 * S22 - S12 * S12;
  float quad  = (S22 * dx * dx - 2.0f * S12 * dx * dy + S11 * dy * dy) / det_s;
  float det_p = P11 * P22 - P12 * P12;
  float det_t = T11 * T22 - T12 * T12;
  float bd = 0.125f * quad
           + 0.5f  * __logf(det_s)
           - 0.25f * (__logf(det_p) + __logf(det_t));
  bd = fminf(fmaxf(bd, -100.0f), 100.0f);
  return fmaxf(1.0f - __expf(-bd), 0.0f);
}

// ---- stage 1: stream tiles via async LDS, compute, per-block partial -------
__global__ void __launch_bounds__(BLOCK)
kf_loss_stage1(const float* __restrict__ pred, const float* __restrict__ targ,
               float* __restrict__ partial, long long N) {
  __shared__ __align__(16) float smem[2 * STAGE_FLTS];  // 40960 B, double buffer
  __shared__ float wred[BLOCK / 32];
  const int tid = threadIdx.x;
  const unsigned ldsBase = (unsigned)(unsigned long long)(void*)smem;
  const long long nTiles = (N + TILE_BOXES - 1) / TILE_BOXES;
  const long long t0 = (long long)blockIdx.x;

  float acc = 0.0f;
  if (t0 < nTiles) stage_tile(pred, targ, t0 * TILE_BOXES, N, ldsBase, tid);

  int buf = 0;
  for (long long t = t0; t < nTiles; t += gridDim.x) {
    const long long tn = t + gridDim.x;
    const bool more = (tn < nTiles);
    if (more)
      stage_tile(pred, targ, tn * TILE_BOXES, N,
                 ldsBase + (unsigned)((buf ^ 1) * STAGE_BYTES), tid);
    if (more) wait_async_le5();   // exactly the 5 just-issued remain in flight
    else      wait_async_0();
    __syncthreads();              // current tile visible to all waves

    const float* P = smem + buf * STAGE_FLTS;
    const float* T = P + TILE_FLTS;
    const long long base = t * TILE_BOXES;
#pragma unroll
    for (int j = 0; j < TILE_BOXES / BLOCK; ++j) {
      const int lb = tid + j * BLOCK;           // stride-5 LDS: conflict-free
      if (base + lb < N) acc += box_loss(P + lb * 5, T + lb * 5);
    }
    __syncthreads();              // done reading buf before it is overwritten
    buf ^= 1;
  }

  // wave32 reduce -> per-wave LDS -> block partial (deterministic)
  for (int o = 16; o > 0; o >>= 1) acc += __shfl_down(acc, o, 32);
  if ((tid & 31) == 0) wred[tid >> 5] = acc;
  __syncthreads();
  if (tid == 0) {
    float s = 0.0f;
#pragma unroll
    for (int w = 0; w < BLOCK / 32; ++w) s += wred[w];
    partial[blockIdx.x] = s;
  }
}

// ---- stage 2: deterministic final reduction of block partials --------------
__global__ void __launch_bounds__(BLOCK)
kf_loss_stage2(const float* __restrict__ partial, float* __restrict__ out, int nb) {
  __shared__ float wred[BLOCK / 32];
  const int tid = threadIdx.x;
  float s = 0.0f;
  for (int i = tid; i < nb; i += BLOCK) s += partial[i];
  for (int o = 16; o > 0; o >>= 1) s += __shfl_down(s, o, 32);
  if ((tid & 31) == 0) wred[tid >> 5] = s;
  __syncthreads();
  if (tid == 0) {
    float r = 0.0f;
#pragma unroll
    for (int w = 0; w < BLOCK / 32; ++w) r += wred[w];
    out[0] = r;
  }
}

extern "C" void kernel_launch(void* const* d_in, const int* in_sizes, int n_in,
                              void* d_out, int out_size, void* d_ws, size_t ws_size,
                              hipStream_t stream) {
  const float* pred = (const float*)d_in[0];
  const float* targ = (const float*)d_in[1];
  const long long N = (long long)in_sizes[0] / 5;   // 4,000,000
  float* partial = (float*)d_ws;                    // NBLOCKS floats (4 KB)
  kf_loss_stage1<<<NBLOCKS, BLOCK, 0, stream>>>(pred, targ, partial, N);
  kf_loss_stage2<<<1, BLOCK, 0, stream>>>(partial, (float*)d_out, NBLOCKS);
}